// EMAVectorQuantizer_9311489098060
// MI455X (gfx1250) — compile-verified
//
#include <hip/hip_runtime.h>
#include <hip/hip_bf16.h>
#include <float.h>

// ---------------------------------------------------------------------------
// Problem constants (from reference):
//   z:      (B=32, D=64, H=64, W=64)  fp32
//   weight: (K=1024, D=64)            fp32
//   N = B*H*W = 131072 flattened vectors, HW = 4096
// Output layout (flat, in return order):
//   [0, 8388608)                       q  (B,D,H,W)
//   [8388608]                          loss = 0.25 * commitment
//   [8388609]                          commitment_loss
//   [8388610, 8389634)                 vq_current_count (K)
//   [8389634, 8455170)                 vq_current_sum   (K,D)
// ---------------------------------------------------------------------------

#define HW      4096
#define DIM     64
#define NK      1024
#define NVEC    131072
#define QELEMS  8388608

typedef __attribute__((ext_vector_type(2))) float v2f;
typedef __attribute__((ext_vector_type(8))) float v8f;

// ---------------------------------------------------------------------------
// Kernel 0: code norms  cnorm[k] = sum_d weight[k][d]^2
// ---------------------------------------------------------------------------
__global__ void vq_cnorm_kernel(const float* __restrict__ w, float* __restrict__ cnorm) {
    int k = blockIdx.x * blockDim.x + threadIdx.x;
    if (k >= NK) return;
    const float* wp = w + k * DIM;
    float s = 0.f;
#pragma unroll
    for (int d = 0; d < DIM; ++d) { float v = wp[d]; s += v * v; }
    cnorm[k] = s;
}

// ---------------------------------------------------------------------------
// Kernel Z: zero the atomic-accumulated output regions + ws scalar
// ---------------------------------------------------------------------------
__global__ void vq_zero_kernel(float* __restrict__ out_tail, float* __restrict__ commit_acc) {
    int i = blockIdx.x * blockDim.x + threadIdx.x;
    const int tail = 2 + NK + NK * DIM; // loss, commit, count, sum
    if (i < tail) out_tail[i] = 0.f;
    if (i == tail) *commit_acc = 0.f;
}

// ---------------------------------------------------------------------------
// Kernel A: argmin over codes via fp32 WMMA 16x16x4.
// Each wave handles 32 z-rows (two 16-row tiles). A (z^T) is held entirely in
// registers; each B (weight) tile load is shared by both row tiles.
// argmin criterion: cnorm[k] - 2 * dot(z, w[k])   (||z||^2 is row-constant).
// ---------------------------------------------------------------------------
__global__ void __launch_bounds__(256)
vq_argmin_kernel(const float* __restrict__ z, const float* __restrict__ w,
                 const float* __restrict__ cnorm, int* __restrict__ idx) {
    const int lane = threadIdx.x & 31;
    const int gw   = blockIdx.x * (blockDim.x >> 5) + (threadIdx.x >> 5);
    const int n0   = gw * 32;              // first of 32 rows owned by this wave
    const int bb   = n0 >> 12;             // batch (HW = 4096)
    const int hw0  = n0 & (HW - 1);
    const int nsub = lane & 15;            // N (code) / M (row) sub-index
    const int koff = (lane >> 4) << 1;     // 0 for lanes 0-15, 2 for lanes 16-31

    // z^T base for this wave: element (row r, feature f) at zp[r + f*HW]
    const float* zp = z + (size_t)bb * (DIM * HW) + hw0 + nsub;

    // Preload all A operands: a[t][ks] covers rows {16t + nsub}, K = {4ks+koff, +1}
    v2f a[2][16];
#pragma unroll
    for (int t = 0; t < 2; ++t) {
#pragma unroll
        for (int ks = 0; ks < 16; ++ks) {
            const float* p = zp + 16 * t + (size_t)(4 * ks + koff) * HW;
            v2f av; av.x = p[0]; av.y = p[HW];
            a[t][ks] = av;
        }
    }

    float minv[2][8];
    int   mini[2][8];
#pragma unroll
    for (int t = 0; t < 2; ++t)
#pragma unroll
        for (int v = 0; v < 8; ++v) { minv[t][v] = FLT_MAX; mini[t][v] = 0; }

    for (int ct = 0; ct < NK / 16; ++ct) {
        const int code0 = ct << 4;
        const float cc = cnorm[code0 + nsub];
        // B tile: element (K=k, N=n) = weight[code0+n][k0+k]; contiguous pair per lane
        const float* wp = w + (size_t)(code0 + nsub) * DIM + koff;

        v8f acc0 = {};
        v8f acc1 = {};
#pragma unroll
        for (int ks = 0; ks < 16; ++ks) {
            v2f bm = *(const v2f*)(wp + 4 * ks);
            acc0 = __builtin_amdgcn_wmma_f32_16x16x4_f32(
                false, a[0][ks], false, bm, (short)0, acc0, false, false);
            acc1 = __builtin_amdgcn_wmma_f32_16x16x4_f32(
                false, a[1][ks], false, bm, (short)0, acc1, false, false);
        }

#pragma unroll
        for (int v = 0; v < 8; ++v) {
            float d0 = __builtin_fmaf(-2.f, acc0[v], cc);
            if (d0 < minv[0][v]) { minv[0][v] = d0; mini[0][v] = code0 + nsub; }
            float d1 = __builtin_fmaf(-2.f, acc1[v], cc);
            if (d1 < minv[1][v]) { minv[1][v] = d1; mini[1][v] = code0 + nsub; }
        }
    }

    // Cross-lane min+argmin butterfly within each 16-lane half (masks 1,2,4,8
    // never cross the half boundary). Tie-break: lower code index (jnp.argmin).
#pragma unroll
    for (int t = 0; t < 2; ++t) {
#pragma unroll
        for (int v = 0; v < 8; ++v) {
            float mv = minv[t][v];
            int   mi = mini[t][v];
#pragma unroll
            for (int m = 1; m <= 8; m <<= 1) {
                float ov = __shfl_xor(mv, m, 32);
                int   oi = __shfl_xor(mi, m, 32);
                if (ov < mv || (ov == mv && oi < mi)) { mv = ov; mi = oi; }
            }
            // C layout: lanes 0-15 hold rows M=v, lanes 16-31 hold rows M=8+v
            if (lane == 0)       idx[n0 + 16 * t + v]     = mi;
            else if (lane == 16) idx[n0 + 16 * t + 8 + v] = mi;
        }
    }
}

// ---------------------------------------------------------------------------
// Kernel B: fused gather / straight-through q / commitment loss / segment sum.
// t is the linear index into q's (B,D,H,W) layout -> coalesced z read + q write.
// ---------------------------------------------------------------------------
__global__ void __launch_bounds__(256)
vq_apply_kernel(const float* __restrict__ z, const float* __restrict__ w,
                const int* __restrict__ idx, float* __restrict__ q,
                float* __restrict__ vqsum, float* __restrict__ commit_acc) {
    const int t  = blockIdx.x * blockDim.x + threadIdx.x;   // exact grid
    const int hw = t & (HW - 1);
    const int d  = (t >> 12) & (DIM - 1);
    const int bb = t >> 18;                                  // DIM*HW = 2^18
    const int n  = (bb << 12) + hw;

    const float zv = z[t];
    const int   c  = idx[n];
    const float wv = w[c * DIM + d];

    q[t] = zv + (wv - zv);                 // straight-through forward value
    const float diff = zv - wv;
    atomicAdd(&vqsum[c * DIM + d], zv);    // vq_current_sum scatter

    __shared__ float red[256];
    red[threadIdx.x] = diff * diff;
    __syncthreads();
#pragma unroll
    for (int s = 128; s > 0; s >>= 1) {
        if (threadIdx.x < s) red[threadIdx.x] += red[threadIdx.x + s];
        __syncthreads();
    }
    if (threadIdx.x == 0) atomicAdd(commit_acc, red[0]);
}

// ---------------------------------------------------------------------------
// Kernel C: code usage histogram (float counts)
// ---------------------------------------------------------------------------
__global__ void vq_count_kernel(const int* __restrict__ idx, float* __restrict__ count) {
    int n = blockIdx.x * blockDim.x + threadIdx.x;           // exact grid
    atomicAdd(&count[idx[n]], 1.0f);
}

// ---------------------------------------------------------------------------
// Kernel F: scalars
// ---------------------------------------------------------------------------
__global__ void vq_finalize_kernel(const float* __restrict__ commit_acc,
                                   float* __restrict__ out_scalars) {
    if (threadIdx.x == 0 && blockIdx.x == 0) {
        float cm = *commit_acc * (1.0f / (float)QELEMS);
        out_scalars[0] = 0.25f * cm;   // loss = BETA * commitment
        out_scalars[1] = cm;           // commitment_loss
    }
}

// ---------------------------------------------------------------------------
extern "C" void kernel_launch(void* const* d_in, const int* in_sizes, int n_in,
                              void* d_out, int out_size, void* d_ws, size_t ws_size,
                              hipStream_t stream) {
    const float* z = (const float*)d_in[0];   // (32,64,64,64)
    const float* w = (const float*)d_in[1];   // (1024,64)
    float* out = (float*)d_out;

    float* q           = out;                        // 8388608
    float* out_scalars = out + QELEMS;               // loss, commitment
    float* count       = out + QELEMS + 2;           // 1024
    float* vqsum       = out + QELEMS + 2 + NK;      // 65536

    // Workspace: [int idx NVEC][float cnorm NK][float commit_acc]
    int*   idx        = (int*)d_ws;
    float* cnorm      = (float*)((char*)d_ws + (size_t)NVEC * 4);
    float* commit_acc = (float*)((char*)d_ws + (size_t)NVEC * 4 + (size_t)NK * 4);

    // 0) code norms
    vq_cnorm_kernel<<<(NK + 255) / 256, 256, 0, stream>>>(w, cnorm);
    // 1) zero atomic targets
    {
        const int tail = 2 + NK + NK * DIM;
        vq_zero_kernel<<<(tail + 1 + 255) / 256, 256, 0, stream>>>(out_scalars, commit_acc);
    }
    // 2) WMMA argmin: 4096 waves * 32 rows = 131072 rows; 8 waves / block
    vq_argmin_kernel<<<(NVEC / 32) / 8, 256, 0, stream>>>(z, w, cnorm, idx);
    // 3) fused gather / q / commit / segment-sum
    vq_apply_kernel<<<QELEMS / 256, 256, 0, stream>>>(z, w, idx, q, vqsum, commit_acc);
    // 4) histogram
    vq_count_kernel<<<NVEC / 256, 256, 0, stream>>>(idx, count);
    // 5) scalars
    vq_finalize_kernel<<<1, 64, 0, stream>>>(commit_acc, out_scalars);
}